// GNNStack_38027640439139
// MI455X (gfx1250) — compile-verified
//
#include <hip/hip_runtime.h>
#include <math.h>

typedef float v2f __attribute__((ext_vector_type(2)));
typedef float v8f __attribute__((ext_vector_type(8)));
typedef unsigned int u32;
typedef u32 tdg0_t __attribute__((ext_vector_type(4)));
typedef int tdg1_t __attribute__((ext_vector_type(8)));

#define HID     128
#define LIN_DIM 328
#define OUTC    5

// TDM-padded LDS pitches (pad inserted by the Tensor Data Mover):
//  - 328-wide tile: pad 4 dwords every 8 dwords   -> off(k) = k + 4*(k>>3), pitch 492
//    banks (44*r) mod 64 distinct for r=0..15, the two half-wave column sets disjoint.
//  - 128-wide tile: pad 4 dwords every 128 dwords -> off(k) = k + 4*(k>>7) = 132*r + c
#define A328_LDS_DW (16 * LIN_DIM + 4 * (16 * LIN_DIM / 8))   // 7872 dwords
#define A128_LDS_DW (16 * 132)                                // 2112 dwords

// ------------------------------------------------------------------ TDM helper
// 2-D tensor_load_to_lds, data_size=4B. Uses the 2-SGPR-group form (VADDR2/3=NULL),
// issued once per wave (caller must be inside a wave-uniform branch).
// pad_interval/pad_amount are the raw D# field codes (interval: 2=8dw, 6=128dw;
// amount: 3 = 4 dwords).
__device__ __forceinline__ void tdm_load_2d(const void* gptr, void* lptr,
                                            u32 row_elems, u32 tile_rows,
                                            u32 tensor_rows, u32 pad_interval,
                                            u32 pad_amount) {
  unsigned long long ga = (unsigned long long)gptr;
  u32 la = (u32)(unsigned long long)lptr;  // low 32 bits of LDS aperture addr
  tdg0_t g0;
  g0[0] = 1u;                                            // count=1, user desc
  g0[1] = la;                                            // lds_addr
  g0[2] = (u32)ga;                                       // global_addr[31:0]
  g0[3] = (u32)((ga >> 32) & 0x1ffffffu) | (2u << 30);   // addr[56:32] | type=2
  tdg1_t g1;
  u32 w0 = (2u << 16)            // data_size = 4 bytes
         | (1u << 20)            // pad_enable
         | (pad_interval << 22)  // pad_interval field
         | (pad_amount << 25);   // pad_amount field
  g1[0] = (int)w0;                                        // workgroup_mask=0
  g1[1] = (int)((row_elems & 0xffffu) << 16);             // tensor_dim0[15:0]
  g1[2] = (int)(((row_elems >> 16) & 0xffffu) |
                ((tensor_rows & 0xffffu) << 16));         // dim0 hi | tensor_dim1 lo
  g1[3] = (int)(((tensor_rows >> 16) & 0xffffu) |
                ((row_elems & 0xffffu) << 16));           // dim1 hi | tile_dim0
  g1[4] = (int)(tile_rows & 0xffffu);                     // tile_dim1 (tile_dim2=0)
  g1[5] = (int)row_elems;                                 // tensor_dim0_stride[31:0]
  g1[6] = 0;
  g1[7] = 0;
  asm volatile("tensor_load_to_lds %0, %1" ::"s"(g0), "s"(g1) : "memory");
}

// ---------------------------------------------------------------- utilities
__global__ void fill_zero(float* __restrict__ p, long long n) {
  long long i = (long long)blockIdx.x * blockDim.x + threadIdx.x;
  long long stride = (long long)gridDim.x * blockDim.x;
  for (; i < n; i += stride) p[i] = 0.0f;
}

// ------------------------------------------------- conv(4x11x11 -> 4x9x9) + bd + relu
__global__ void conv_act_kernel(const float* __restrict__ x,
                                const float* __restrict__ bd,
                                const float* __restrict__ w,
                                const float* __restrict__ b,
                                float* __restrict__ act, int N) {
  __shared__ float xs[4 * 121];
  int node = blockIdx.x;
  int tid  = threadIdx.x;
  const float* xp = x + (long long)node * (4 * 121);
  for (int i = tid; i < 4 * 121; i += blockDim.x) xs[i] = xp[i];
  __syncthreads();
  float* ap = act + (long long)node * LIN_DIM;
  for (int idx = tid; idx < 324; idx += blockDim.x) {
    int oc = idx / 81, p = idx % 81;
    int oy = p / 9, ox = p % 9;
    float sum = b[oc];
    const float* wp = w + oc * 36;
#pragma unroll
    for (int ic = 0; ic < 4; ++ic)
#pragma unroll
      for (int ky = 0; ky < 3; ++ky)
#pragma unroll
        for (int kx = 0; kx < 3; ++kx)
          sum += xs[ic * 121 + (oy + ky) * 11 + (ox + kx)] * wp[ic * 9 + ky * 3 + kx];
    ap[idx] = fmaxf(sum, 0.0f);
  }
  if (tid < 4) ap[324 + tid] = fmaxf(bd[(long long)node * 4 + tid], 0.0f);
}

// ------------------------- act[N,328] @ {lin_self_w, lin_w}^T + bias  (f32 WMMA)
// 512 threads = 16 waves: waves 0-7 -> self matrix, waves 8-15 -> neighbor matrix.
// A tile staged by the Tensor Data Mover (pad 4dw/8dw -> conflict-free pitch 492).
__global__ void gemm328_kernel(const float* __restrict__ act,
                               const float* __restrict__ w_self,
                               const float* __restrict__ b_self,
                               const float* __restrict__ w_nb,
                               const float* __restrict__ b_nb,
                               float* __restrict__ out_self,
                               float* __restrict__ out_nb, int N) {
  __shared__ float As[A328_LDS_DW];
  int row0 = blockIdx.x * 16;
  int tid  = threadIdx.x;
  if ((tid >> 5) == 0) {  // wave-uniform: wave 0 issues the DMA
    tdm_load_2d(act + (size_t)row0 * LIN_DIM, As, LIN_DIM, 16,
                (u32)(N - row0), /*interval 8dw*/ 2, /*amount 4dw*/ 3);
    __builtin_amdgcn_s_wait_tensorcnt(0);
  }
  __syncthreads();

  int wave = tid >> 5, lane = tid & 31;
  int matsel  = wave >> 3;
  int colbase = (wave & 7) * 16;
  const float* W = matsel ? w_nb : w_self;
  const float* B = matsel ? b_nb : b_self;
  float* O       = matsel ? out_nb : out_self;
  int arow = lane & 15;
  int koff = (lane >> 4) * 2;
  int n    = colbase + (lane & 15);
  const float* wrow = W + (long long)n * LIN_DIM;  // B[k][n] = W[n][k]
  __builtin_prefetch(wrow, 0, 3);
  v8f c = {};
  for (int kk = 0; kk < LIN_DIM; kk += 4) {
    int k = arow * LIN_DIM + kk + koff;
    int off = k + 4 * (k >> 3);  // TDM pad mapping (pair stays adjacent)
    v2f a, bb;
    a.x  = As[off];
    a.y  = As[off + 1];
    bb.x = wrow[kk + koff];
    bb.y = wrow[kk + koff + 1];
    c = __builtin_amdgcn_wmma_f32_16x16x4_f32(false, a, false, bb, (short)0, c,
                                              false, false);
  }
  float bias = B[n];
  if (row0 + 16 <= N) {  // block-uniform full-tile fast path
#pragma unroll
    for (int v = 0; v < 8; ++v) {
      int mrow = v + (lane >> 4) * 8;
      O[(long long)(row0 + mrow) * HID + n] = c[v] + bias;
    }
  } else {
#pragma unroll
    for (int v = 0; v < 8; ++v) {
      int mrow = v + (lane >> 4) * 8;
      if (row0 + mrow < N) O[(long long)(row0 + mrow) * HID + n] = c[v] + bias;
    }
  }
}

// ---- out = (A1/deg?) @ W1^T + bias1 [+ A2 @ W2^T]   (128x128, f32 WMMA)
// Separate accumulators; 1/deg applied once per output row in the epilogue.
__global__ void gemm128_kernel(const float* __restrict__ A1,
                               const float* __restrict__ deg,  // null -> no mean scaling
                               const float* __restrict__ W1,
                               const float* __restrict__ bias1,
                               const float* __restrict__ A2,   // null -> single GEMM
                               const float* __restrict__ W2,
                               float* __restrict__ out, int N) {
  __shared__ float A1s[A128_LDS_DW];
  __shared__ float A2s[A128_LDS_DW];
  int row0 = blockIdx.x * 16;
  int tid  = threadIdx.x;
  if ((tid >> 5) == 0) {  // wave-uniform: wave 0 issues the DMA(s)
    tdm_load_2d(A1 + (size_t)row0 * HID, A1s, HID, 16, (u32)(N - row0),
                /*interval 128dw*/ 6, /*amount 4dw*/ 3);
    if (A2)
      tdm_load_2d(A2 + (size_t)row0 * HID, A2s, HID, 16, (u32)(N - row0),
                  /*interval 128dw*/ 6, /*amount 4dw*/ 3);
    __builtin_amdgcn_s_wait_tensorcnt(0);
  }
  __syncthreads();

  int wave = tid >> 5, lane = tid & 31;
  int colbase = wave * 16;
  int arow = lane & 15, koff = (lane >> 4) * 2;
  int n = colbase + (lane & 15);
  const float* w1row = W1 + (long long)n * HID;
  const float* w2row = A2 ? (W2 + (long long)n * HID) : nullptr;
  __builtin_prefetch(w1row, 0, 3);
  v8f c1 = {};
  v8f c2 = {};
  for (int kk = 0; kk < HID; kk += 4) {
    int off = arow * 132 + kk + koff;  // TDM pad -> pitch 132
    v2f a, bvec;
    a.x = A1s[off];
    a.y = A1s[off + 1];
    bvec.x = w1row[kk + koff];
    bvec.y = w1row[kk + koff + 1];
    c1 = __builtin_amdgcn_wmma_f32_16x16x4_f32(false, a, false, bvec, (short)0, c1,
                                               false, false);
    if (w2row) {
      v2f a2, b2;
      a2.x = A2s[off];
      a2.y = A2s[off + 1];
      b2.x = w2row[kk + koff];
      b2.y = w2row[kk + koff + 1];
      c2 = __builtin_amdgcn_wmma_f32_16x16x4_f32(false, a2, false, b2, (short)0, c2,
                                                 false, false);
    }
  }
  float bv = bias1[n];
  if (row0 + 16 <= N) {
#pragma unroll
    for (int v = 0; v < 8; ++v) {
      int mrow = v + (lane >> 4) * 8;
      float rd = deg ? (1.0f / fmaxf(deg[row0 + mrow], 1.0f)) : 1.0f;
      out[(long long)(row0 + mrow) * HID + n] = c1[v] * rd + c2[v] + bv;
    }
  } else {
#pragma unroll
    for (int v = 0; v < 8; ++v) {
      int mrow = v + (lane >> 4) * 8;
      if (row0 + mrow < N) {
        float rd = deg ? (1.0f / fmaxf(deg[row0 + mrow], 1.0f)) : 1.0f;
        out[(long long)(row0 + mrow) * HID + n] = c1[v] * rd + c2[v] + bv;
      }
    }
  }
}

// ------------- layer-0 edge pass: accum[dst] += neigh[src]*(src!=dst); deg[dst] += 1
__global__ void edge_msg_kernel(const int* __restrict__ src,
                                const int* __restrict__ dst,
                                const float* __restrict__ neigh,
                                float* __restrict__ accum,
                                float* __restrict__ deg, int E) {
  int gw   = (blockIdx.x * blockDim.x + threadIdx.x) >> 5;
  int lane = threadIdx.x & 31;
  if (gw >= E) return;
  int s = src[gw], d = dst[gw];
  if (lane == 0) atomicAdd(&deg[d], 1.0f);
  if (s == d) return;  // self-loop messages are masked in layer 0
  const float4* np = (const float4*)(neigh + (long long)s * HID);
  float4 v  = np[lane];
  float* op = accum + (long long)d * HID + lane * 4;
  atomicAdd(op + 0, v.x);
  atomicAdd(op + 1, v.y);
  atomicAdd(op + 2, v.z);
  atomicAdd(op + 3, v.w);
}

// ------------- SAGE edge pass (no mask): m[dst] += xh[src]
__global__ void sage_scatter_kernel(const int* __restrict__ src,
                                    const int* __restrict__ dst,
                                    const float* __restrict__ xh,
                                    float* __restrict__ m, int E) {
  int gw   = (blockIdx.x * blockDim.x + threadIdx.x) >> 5;
  int lane = threadIdx.x & 31;
  if (gw >= E) return;
  int s = src[gw], d = dst[gw];
  const float4* np = (const float4*)(xh + (long long)s * HID);
  float4 v  = np[lane];
  float* op = m + (long long)d * HID + lane * 4;
  atomicAdd(op + 0, v.x);
  atomicAdd(op + 1, v.y);
  atomicAdd(op + 2, v.z);
  atomicAdd(op + 3, v.w);
}

// ------------- relu + optional LayerNorm over 128 features (one block per row)
__global__ void act_ln_kernel(const float* __restrict__ pre,
                              const float* __restrict__ gamma,  // null -> no LN
                              const float* __restrict__ beta,
                              float* __restrict__ out, int N) {
  __shared__ float s1[128];
  __shared__ float s2[128];
  int row = blockIdx.x;
  int tid = threadIdx.x;
  float v = fmaxf(pre[(long long)row * HID + tid], 0.0f);
  if (!gamma) {  // uniform branch (kernel arg)
    out[(long long)row * HID + tid] = v;
    return;
  }
  s1[tid] = v;
  s2[tid] = v * v;
  __syncthreads();
  for (int off = 64; off > 0; off >>= 1) {
    if (tid < off) { s1[tid] += s1[tid + off]; s2[tid] += s2[tid + off]; }
    __syncthreads();
  }
  float mean = s1[0] * (1.0f / 128.0f);
  float var  = s2[0] * (1.0f / 128.0f) - mean * mean;
  float inv  = rsqrtf(var + 1e-5f);
  out[(long long)row * HID + tid] = (v - mean) * inv * gamma[tid] + beta[tid];
}

// ------------- z2 = z1 @ post_w2^T + b2 ; log_softmax (one wave32 per row)
__global__ void post2_kernel(const float* __restrict__ z1,
                             const float* __restrict__ w2,
                             const float* __restrict__ b2,
                             float* __restrict__ out, int N) {
  int gw   = (blockIdx.x * blockDim.x + threadIdx.x) >> 5;
  int lane = threadIdx.x & 31;
  if (gw >= N) return;
  const float4* xp = (const float4*)(z1 + (long long)gw * HID);
  float4 xv = xp[lane];
  float z[OUTC];
#pragma unroll
  for (int o = 0; o < OUTC; ++o) {
    const float4* wp = (const float4*)(w2 + o * HID);
    float4 wv = wp[lane];
    float p = xv.x * wv.x + xv.y * wv.y + xv.z * wv.z + xv.w * wv.w;
#pragma unroll
    for (int off = 16; off > 0; off >>= 1) p += __shfl_down(p, off, 32);
    z[o] = p + b2[o];  // valid on lane 0
  }
  if (lane == 0) {
    float mx = z[0];
#pragma unroll
    for (int o = 1; o < OUTC; ++o) mx = fmaxf(mx, z[o]);
    float se = 0.0f;
#pragma unroll
    for (int o = 0; o < OUTC; ++o) se += expf(z[o] - mx);
    float lse = logf(se);
    float* op = out + (long long)gw * OUTC;
#pragma unroll
    for (int o = 0; o < OUTC; ++o) op[o] = z[o] - mx - lse;
  }
}

// ---------------------------------------------------------------- driver
extern "C" void kernel_launch(void* const* d_in, const int* in_sizes, int n_in,
                              void* d_out, int out_size, void* d_ws, size_t ws_size,
                              hipStream_t stream) {
  const float* x          = (const float*)d_in[0];
  const float* bd         = (const float*)d_in[1];
  const int*   ei         = (const int*)  d_in[2];
  const float* conv_w     = (const float*)d_in[3];
  const float* conv_b     = (const float*)d_in[4];
  const float* lin_self_w = (const float*)d_in[5];
  const float* lin_self_b = (const float*)d_in[6];
  const float* lin_w      = (const float*)d_in[7];
  const float* lin_b      = (const float*)d_in[8];
  const float* sage_l_w   = (const float*)d_in[9];
  const float* sage_l_b   = (const float*)d_in[10];
  const float* sage_r_w   = (const float*)d_in[11];
  const float* ln_g       = (const float*)d_in[12];
  const float* ln_bt      = (const float*)d_in[13];
  const float* post_w1    = (const float*)d_in[14];
  const float* post_b1    = (const float*)d_in[15];
  const float* post_w2    = (const float*)d_in[16];
  const float* post_b2    = (const float*)d_in[17];

  int N = in_sizes[1] / 4;   // bd_pred is [N,4]
  int E = in_sizes[2] / 2;   // edge_index is [2,E]
  const int* src = ei;
  const int* dst = ei + E;

  // workspace layout (~143 MB): act | xh | mb | pre | deg
  float* ws  = (float*)d_ws;
  float* act = ws;                          // N*328
  float* xh  = act + (size_t)N * LIN_DIM;   // N*128  running features
  float* mb  = xh  + (size_t)N * HID;       // N*128  neigh / scatter accum / z1
  float* pre = mb  + (size_t)N * HID;       // N*128  pre-activation
  float* deg = pre + (size_t)N * HID;       // N

  float* emb_out = (float*)d_out;                     // [N,128]
  float* lp_out  = emb_out + (size_t)N * HID;         // [N,5]

  int nt16 = (N + 15) / 16;
  int eblk = (E * 32 + 255) / 256;  // one wave per edge, 8 waves/block

  // layer 0: conv + act, self/neigh GEMMs, masked edge aggregation + degree
  conv_act_kernel<<<N, 128, 0, stream>>>(x, bd, conv_w, conv_b, act, N);
  fill_zero<<<256, 256, 0, stream>>>(deg, N);
  gemm328_kernel<<<nt16, 512, 0, stream>>>(act, lin_self_w, lin_self_b, lin_w,
                                           lin_b, xh, mb, N);
  edge_msg_kernel<<<eblk, 256, 0, stream>>>(src, dst, mb, xh, deg, E);

  // 3 SAGE layers
  for (int i = 0; i < 3; ++i) {
    fill_zero<<<1024, 256, 0, stream>>>(mb, (long long)N * HID);
    sage_scatter_kernel<<<eblk, 256, 0, stream>>>(src, dst, xh, mb, E);
    float* outbuf = (i == 2) ? emb_out : pre;  // layer-2 pre-act == emb output
    gemm128_kernel<<<nt16, 256, 0, stream>>>(
        mb, deg, sage_l_w + (size_t)i * HID * HID, sage_l_b + (size_t)i * HID,
        xh, sage_r_w + (size_t)i * HID * HID, outbuf, N);
    const float* g = (i < 2) ? (ln_g + (size_t)i * HID) : nullptr;
    const float* b = (i < 2) ? (ln_bt + (size_t)i * HID) : nullptr;
    act_ln_kernel<<<N, 128, 0, stream>>>(outbuf, g, b, xh, N);
  }

  // post_mp: z1 = xh@W1^T+b1 (WMMA), then z2 + log_softmax
  gemm128_kernel<<<nt16, 256, 0, stream>>>(xh, nullptr, post_w1, post_b1,
                                           nullptr, nullptr, mb, N);
  post2_kernel<<<(N + 7) / 8, 256, 0, stream>>>(mb, post_w2, post_b2, lp_out, N);
}